// Model_18726057411287
// MI455X (gfx1250) — compile-verified
//
#include <hip/hip_runtime.h>
#include <stdint.h>

// ---------------- problem constants ----------------
#define VOCAB 128000
#define TOPK  50
#define BINS  2048              // histogram over top 11 bits of monotonic key
#define CAP   2048              // per-row candidate capacity (threshold bin)
#define TPB   256               // 8 waves (wave32)
#define TILES (VOCAB / 4 / TPB) // 125 float4 tiles per row
#define CHUNK (BINS / TPB)      // 8 bins per thread for the threshold scan

// workspace layout per row, in uint64 units: header(4) | hi(64) | cand(CAP)
#define HDR_U64 4
#define HI_U64  64
#define ROW_U64 (HDR_U64 + HI_U64 + CAP)

// ---------------- monotonic float<->uint key ----------------
__device__ __forceinline__ uint32_t f2k(float f) {
    uint32_t u = __float_as_uint(f);
    return u ^ ((u >> 31) ? 0xFFFFFFFFu : 0x80000000u);
}
__device__ __forceinline__ float k2f(uint32_t k) {
    uint32_t u = (k >> 31) ? (k ^ 0x80000000u) : ~k;
    return __uint_as_float(u);
}

// ---------------- CDNA5 async global->LDS staging ----------------
#if defined(__has_builtin)
#if __has_builtin(__builtin_amdgcn_global_load_async_to_lds_b128)
#define HAVE_ASYNC 1
#endif
#endif
#ifndef HAVE_ASYNC
#define HAVE_ASYNC 0
#endif

__device__ __forceinline__ void wait_async_le1() {
    asm volatile("s_wait_asynccnt 0x1" ::: "memory");
}
__device__ __forceinline__ void wait_async_le0() {
    asm volatile("s_wait_asynccnt 0x0" ::: "memory");
}

#if HAVE_ASYNC
typedef int v4i_t __attribute__((vector_size(16)));
typedef __attribute__((address_space(1))) v4i_t* gv4i_p;
typedef __attribute__((address_space(3))) v4i_t* lv4i_p;
typedef __attribute__((address_space(3))) void*  lvoid_p;

__device__ __forceinline__ void async_ld_b128(const float4* g, float4* lds) {
    __builtin_amdgcn_global_load_async_to_lds_b128(
        (gv4i_p)(uintptr_t)g,
        (lv4i_p)(lvoid_p)lds,
        /*offset=*/0, /*cpol=*/0);
}
#endif

// Stream one row (VOCAB floats) through the functor, float4 at a time.
// Async path: per-lane private LDS slots, double-buffered tiles, waits via
// s_wait_asynccnt. No cross-lane sharing -> no barriers needed in the loop.
template <class OP>
__device__ __forceinline__ void stream_row(const float4* __restrict__ src, OP op) {
    const int tid = threadIdx.x;
#if HAVE_ASYNC
    __shared__ float4 stage[2][TPB];
    async_ld_b128(src + tid, &stage[0][tid]);
    for (int t = 0; t < TILES; ++t) {
        const int nb = t + 1;
        if (nb < TILES) {
            async_ld_b128(src + nb * TPB + tid, &stage[nb & 1][tid]);
            wait_async_le1();   // tile t has landed (in-order completion)
        } else {
            wait_async_le0();
        }
        float4 v = stage[t & 1][tid];
        op(v, (t * TPB + tid) * 4);
    }
#else
    for (int t = 0; t < TILES; ++t) {
        float4 v = src[t * TPB + tid];
        op(v, (t * TPB + tid) * 4);
    }
#endif
}

// ---------------- kernel A: per-row top-50 candidate selection ----------------
__global__ void __launch_bounds__(TPB)
topk_select_kernel(const float* __restrict__ logits, uint64_t* __restrict__ ws) {
    const int row = blockIdx.x;
    const int tid = threadIdx.x;

    __shared__ unsigned hist[BINS];
    __shared__ unsigned csum[TPB];
    __shared__ unsigned s_t, s_chi, cnt_hi, cnt_cand;

    for (int i = tid; i < BINS; i += TPB) hist[i] = 0u;
    if (tid == 0) { cnt_hi = 0u; cnt_cand = 0u; }
    __syncthreads();

    const float4* src = (const float4*)(logits + (size_t)row * VOCAB);

    // pass 1: histogram of key's top 11 bits (LDS ds_add_u32 atomics)
    stream_row(src, [&](float4 v, int /*e*/) {
        atomicAdd(&hist[f2k(v.x) >> 21], 1u);
        atomicAdd(&hist[f2k(v.y) >> 21], 1u);
        atomicAdd(&hist[f2k(v.z) >> 21], 1u);
        atomicAdd(&hist[f2k(v.w) >> 21], 1u);
    });
    __syncthreads();

    // ---- parallel threshold search ----
    // chunk sums: thread c owns bins [c*CHUNK, c*CHUNK+CHUNK)
    {
        unsigned s = 0;
        #pragma unroll
        for (int k = 0; k < CHUNK; ++k) s += hist[tid * CHUNK + k];
        csum[tid] = s;
        __syncthreads();
        // inclusive suffix scan (Hillis-Steele, 8 steps): csum[c] = sum_{c'>=c}
        #pragma unroll
        for (int off = 1; off < TPB; off <<= 1) {
            unsigned v   = csum[tid];
            unsigned add = (tid + off < TPB) ? csum[tid + off] : 0u;
            __syncthreads();
            csum[tid] = v + add;
            __syncthreads();
        }
        // the chunk where the K-th element falls: excl < K <= incl
        unsigned excl = (tid + 1 < TPB) ? csum[tid + 1] : 0u;
        unsigned incl = csum[tid];
        if (excl < TOPK && incl >= TOPK) {
            unsigned acc = excl;
            for (int k = CHUNK - 1; k >= 0; --k) {
                unsigned c = hist[tid * CHUNK + k];
                if (acc + c >= TOPK) { s_t = (unsigned)(tid * CHUNK + k); s_chi = acc; break; }
                acc += c;
            }
        }
    }
    __syncthreads();
    const unsigned t = s_t;

    uint64_t* hi   = ws + (size_t)row * ROW_U64 + HDR_U64;
    uint64_t* cand = hi + HI_U64;

    // pass 2 (L2-resident): collect winners and threshold-bin candidates.
    // pack = key<<32 | ~idx  => uint64 descending order == (value desc, idx asc)
    stream_row(src, [&](float4 v, int e) {
        float xs[4] = {v.x, v.y, v.z, v.w};
        #pragma unroll
        for (int c = 0; c < 4; ++c) {
            uint32_t k = f2k(xs[c]);
            uint32_t bin = k >> 21;
            if (bin > t) {
                unsigned s = atomicAdd(&cnt_hi, 1u);
                if (s < HI_U64)
                    hi[s] = (((uint64_t)k) << 32) | (uint64_t)(uint32_t)~(uint32_t)(e + c);
            } else if (bin == t) {
                unsigned s = atomicAdd(&cnt_cand, 1u);
                if (s < CAP)
                    cand[s] = (((uint64_t)k) << 32) | (uint64_t)(uint32_t)~(uint32_t)(e + c);
            }
        }
    });
    __syncthreads();

    if (tid == 0) {
        unsigned nc = cnt_cand; if (nc > CAP) nc = CAP;
        ws[(size_t)row * ROW_U64] = (((uint64_t)nc) << 32) | (uint64_t)s_chi;
    }
}

// ---------------- Threefry-2x32-20 (JAX), key = (0, 42) ----------------
__device__ __forceinline__ uint2 tf2x32_key42(uint32_t x0, uint32_t x1) {
    const uint32_t ks0 = 0u, ks1 = 42u, ks2 = 0u ^ 42u ^ 0x1BD11BDAu;
    x0 += ks0; x1 += ks1;
#define TFR(r) { x0 += x1; x1 = (x1 << (r)) | (x1 >> (32 - (r))); x1 ^= x0; }
    TFR(13) TFR(15) TFR(26) TFR(6)   x0 += ks1; x1 += ks2 + 1u;
    TFR(17) TFR(29) TFR(16) TFR(24)  x0 += ks2; x1 += ks0 + 2u;
    TFR(13) TFR(15) TFR(26) TFR(6)   x0 += ks0; x1 += ks1 + 3u;
    TFR(17) TFR(29) TFR(16) TFR(24)  x0 += ks1; x1 += ks2 + 4u;
    TFR(13) TFR(15) TFR(26) TFR(6)   x0 += ks2; x1 += ks0 + 5u;
#undef TFR
    return make_uint2(x0, x1);
}

// ---------------- kernel B: sort 50, Gumbel-categorical sample, gather ----------------
__global__ void __launch_bounds__(TPB)
sample_kernel(const uint64_t* __restrict__ ws, int* __restrict__ out, unsigned half) {
    const int row = blockIdx.x;
    const int tid = threadIdx.x;

    const uint64_t* hdr  = ws + (size_t)row * ROW_U64;
    const uint64_t* hi   = hdr + HDR_U64;
    const uint64_t* cand = hi + HI_U64;

    __shared__ uint64_t sc[CAP];
    __shared__ uint64_t pool[TOPK];
    __shared__ uint64_t sorted[TOPK];
    __shared__ uint64_t skey[TOPK];
    __shared__ unsigned s_chi_sh, s_n_sh;

    if (tid == 0) {
        uint64_t h = hdr[0];
        s_chi_sh = (unsigned)(h & 0xffffffffu);
        s_n_sh   = (unsigned)(h >> 32);
    }
    if (tid < TOPK) { pool[tid] = 0; sorted[tid] = 0; }
    __syncthreads();

    const unsigned chi = s_chi_sh;
    unsigned n = s_n_sh; if (n > CAP) n = CAP;
    const int r = TOPK - (int)chi;

    for (unsigned i = tid; i < n; i += TPB) sc[i] = cand[i];
    if (tid < chi) pool[tid] = hi[tid];
    __syncthreads();

    // rank-select top r candidates (keys are unique; O(n^2/TPB), n ~ few hundred)
    for (unsigned i = tid; i < n; i += TPB) {
        uint64_t me = sc[i];
        int rank = 0;
        for (unsigned j = 0; j < n; ++j) rank += (sc[j] > me);
        if (rank < r) pool[chi + (unsigned)rank] = me;
    }
    __syncthreads();

    // full sort of the 50 (value desc, index asc) to match jax.lax.top_k order
    if (tid < TOPK) {
        uint64_t me = pool[tid];
        int rank = 0;
        for (int j = 0; j < TOPK; ++j) rank += (pool[j] > me);
        sorted[rank] = me;
    }
    __syncthreads();

    // Gumbel noise per sorted slot, JAX threefry counter layout
    if (tid < TOPK) {
        uint64_t e = sorted[tid];
        float v = k2f((uint32_t)(e >> 32));
        unsigned j = (unsigned)row * TOPK + (unsigned)tid;
        uint32_t bits;
        if (j < half) { uint2 o = tf2x32_key42(j, j + half); bits = o.x; }
        else          { uint2 o = tf2x32_key42(j - half, j); bits = o.y; }
        float f = __uint_as_float((bits >> 9) | 0x3f800000u) - 1.0f;  // [0,1)
        const float TINYF = 1.1754943508222875e-38f;
        float u = fmaxf(TINYF, f * (1.0f - TINYF) + TINYF);
        float g = -logf(-logf(u));
        float s = v + g;
        // argmax with first-wins ties: maximize (key(s), ~slot)
        skey[tid] = (((uint64_t)f2k(s)) << 32) | (uint64_t)(uint32_t)~(uint32_t)tid;
    }
    __syncthreads();

    if (tid == 0) {
        uint64_t best = skey[0];
        for (int j = 1; j < TOPK; ++j) if (skey[j] > best) best = skey[j];
        unsigned slot = ~(uint32_t)best;         // winning sorted slot
        unsigned idx  = ~(uint32_t)sorted[slot]; // original vocab index
        out[row] = (int)idx;
    }
}

// ---------------- host entry ----------------
extern "C" void kernel_launch(void* const* d_in, const int* in_sizes, int n_in,
                              void* d_out, int out_size, void* d_ws, size_t ws_size,
                              hipStream_t stream) {
    (void)n_in; (void)out_size; (void)ws_size;
    const float* logits = (const float*)d_in[0];
    const int B = in_sizes[0] / VOCAB;            // 256
    uint64_t* ws = (uint64_t*)d_ws;
    int* out = (int*)d_out;

    topk_select_kernel<<<B, TPB, 0, stream>>>(logits, ws);
    sample_kernel<<<B, TPB, 0, stream>>>(ws, out, (unsigned)((B * TOPK) / 2));
}